// PowerVector_5446018531796
// MI455X (gfx1250) — compile-verified
//
#include <hip/hip_runtime.h>

typedef __attribute__((ext_vector_type(2))) float v2f;
typedef __attribute__((ext_vector_type(8))) float v8f;

// f32 WMMA: NEG/NEG_HI must be zero for F32 (only CNeg allowed per ISA), so
// all modifier immediates are 0; sign handling is done in VALU on the A frag.
#define WMMA4(A, B, C) \
  __builtin_amdgcn_wmma_f32_16x16x4_f32(false, (A), false, (B), (short)0, (C), false, false)

// Constants
// B=2 T=400 C=8 F=512 K=40 (padded 48) C2=64 P=64
#define NBT   800        // B*T
#define NTOT  2048000    // B*T*K*C2 (one component)

// ---------------------------------------------------------------------------
// Kernel 1: per-(b,t) fused cov -> phase-adjust -> band projection (complex
// GEMM [64x512]x[512x40] via V_WMMA_F32_16X16X4_F32) -> diag-normalize.
// Block = 384 threads = 12 waves = 4 M-tiles x 3 N-tiles.
// ---------------------------------------------------------------------------
__global__ __launch_bounds__(384) void pv_cov_band_kernel(
    const float* __restrict__ bins_r, const float* __restrict__ bins_i,
    const float* __restrict__ bm_r,   const float* __restrict__ bm_i,
    float* __restrict__ ws)
{
  extern __shared__ float smem[];
  float* xs_r  = smem;             // 4096  (8 ch x 512 f)
  float* xs_i  = xs_r  + 4096;     // 4096
  float* invtr = xs_i  + 4096;     // 512
  float* adj_r = invtr + 512;      // 4160  (64 c2 x 64 f, stride 65)
  float* adj_i = adj_r + 4160;     // 4160
  float* bmc_r = adj_i + 4160;     // 3072  (64 f x 48 bands)
  float* bmc_i = bmc_r + 3072;     // 3072
  float* dsum  = bmc_i + 3072;     // 64
  // total 23232 floats = 92928 bytes dynamic LDS (WGP has 320 KB)

  const int tid  = threadIdx.x;
  const int bt   = blockIdx.x;      // b*400 + t
  const int lane = tid & 31;
  const int wid  = tid >> 5;        // 0..11
  const int mt   = wid / 3;         // c2 tile 0..3
  const int nt   = wid % 3;         // band tile 0..2
  const int ll   = lane & 15;
  const int hi   = lane >> 4;

  // ---- load bins for this (b,t): (8,512) complex ----
  {
    const size_t base = (size_t)bt * 4096;
    for (int e = tid; e < 4096; e += 384) {
      xs_r[e] = bins_r[base + e];
      xs_i[e] = bins_i[base + e];
    }
  }
  __syncthreads();

  // ---- inverse clipped trace per bin ----
  for (int f = tid; f < 512; f += 384) {
    float tr = 0.f;
    #pragma unroll
    for (int ch = 0; ch < 8; ++ch) {
      float a = xs_r[ch*512 + f], b = xs_i[ch*512 + f];
      tr += a*a + b*b;
    }
    invtr[f] = 1.f / fmaxf(tr, 1e-20f);
  }
  __syncthreads();

  v8f accR = {};   // real part of band_cov tile
  v8f accI = {};   // imag part

  for (int kc = 0; kc < 8; ++kc) {      // K chunks of 64 bins
    const int f0 = kc * 64;

    // ---- build phase-adjusted covariance chunk: adj[c2][flo] ----
    for (int e = tid; e < 4096; e += 384) {
      int c2  = e >> 6;
      int flo = e & 63;
      int f   = f0 + flo;
      int i   = c2 >> 3, j = c2 & 7;
      float xr = xs_r[i*512 + f], xi = xs_i[i*512 + f];
      float yr = xs_r[j*512 + f], yi = xs_i[j*512 + f];
      float it = invtr[f];
      float cre = (xr*yr + xi*yi) * it;     // x_i * conj(x_j) / trace
      float cim = (xi*yr - xr*yi) * it;
      float ar, ai;
      if (i == j) {
        ar = cre; ai = 0.f;                 // diagonals keep original value
      } else {
        int fl = f - 1; fl = fl < 0 ? 0 : (fl > 509 ? 509 : fl);
        int fr = fl + 2;
        float lxr = xs_r[i*512 + fl], lxi = xs_i[i*512 + fl];
        float lyr = xs_r[j*512 + fl], lyi = xs_i[j*512 + fl];
        float clr = lxr*lyr + lxi*lyi;      // c[fl] (un-normalized; angle-invariant)
        float cli = lxi*lyr - lxr*lyi;
        float rxr = xs_r[i*512 + fr], rxi = xs_i[i*512 + fr];
        float ryr = xs_r[j*512 + fr], ryi = xs_i[j*512 + fr];
        float crr = rxr*ryr + rxi*ryi;      // c[fr]
        float cri = rxi*ryr - rxr*ryi;
        float pr  = clr*crr + cli*cri;      // conj(cl) * cr
        float pi  = clr*cri - cli*crr;
        float mag = sqrtf(cre*cre + cim*cim);
        float n2  = pr*pr + pi*pi;
        if (n2 > 0.f) {                     // |c| * exp(i*atan2(pi,pr)) == |c|*p/|p|
          float s = mag * rsqrtf(n2);
          ar = pr * s; ai = pi * s;
        } else {                            // atan2(0,0) == 0 -> unit (1,0)
          ar = mag; ai = 0.f;
        }
      }
      adj_r[c2*65 + flo] = ar;
      adj_i[c2*65 + flo] = ai;
    }

    // ---- stage band-matrix chunk (64 bins x 48 bands, zero-padded) ----
    for (int e = tid; e < 3072; e += 384) {
      int flo = e / 48, nb = e - flo*48;
      float vr = 0.f, vi = 0.f;
      if (nb < 40) {
        vr = bm_r[(f0 + flo)*40 + nb];
        vi = bm_i[(f0 + flo)*40 + nb];
      }
      bmc_r[flo*48 + nb] = vr;
      bmc_i[flo*48 + nb] = vi;
    }
    __syncthreads();

    // ---- complex GEMM tile: K=64 in steps of 4, 4 WMMAs per step ----
    const int mrow = mt*16 + ll;
    const int ncol = nt*16 + ll;
    #pragma unroll 4
    for (int kk = 0; kk < 16; ++kk) {
      int kb = kk*4 + hi*2;
      v2f Ar, Ai, Br, Bi;
      Ar.x = adj_r[mrow*65 + kb];   Ar.y = adj_r[mrow*65 + kb + 1];
      Ai.x = adj_i[mrow*65 + kb];   Ai.y = adj_i[mrow*65 + kb + 1];
      Br.x = bmc_r[kb*48 + ncol];   Br.y = bmc_r[(kb+1)*48 + ncol];
      Bi.x = bmc_i[kb*48 + ncol];   Bi.y = bmc_i[(kb+1)*48 + ncol];
      v2f nAi; nAi.x = -Ai.x; nAi.y = -Ai.y;   // VALU negate (f32 WMMA has no neg_a)
      accR = WMMA4(Ar,  Br, accR);   // + Ar*Br
      accR = WMMA4(nAi, Bi, accR);   // - Ai*Bi
      accI = WMMA4(Ar,  Bi, accI);   // + Ar*Bi
      accI = WMMA4(Ai,  Br, accI);   // + Ai*Br
    }
    __syncthreads();
  }

  // ---- stage accumulators to LDS (reuse adj space): st[c2*48 + band] ----
  float* st_r = adj_r;
  float* st_i = adj_i;
  #pragma unroll
  for (int r = 0; r < 8; ++r) {
    int row = mt*16 + r + hi*8;
    int col = nt*16 + ll;
    st_r[row*48 + col] = accR[r];
    st_i[row*48 + col] = accI[r];
  }
  __syncthreads();

  // ---- per-band inverse diagonal sum (diag c2 = i*9) ----
  for (int k = tid; k < 48; k += 384) {
    float s = 0.f;
    #pragma unroll
    for (int d = 0; d < 8; ++d) s += st_r[(d*9)*48 + k];
    dsum[k] = 1.f / fmaxf(s, 1e-20f);
  }
  __syncthreads();

  // ---- write normalized band covariance: ws = [re | im], (bt*40 + k)*64 + c ----
  {
    float* ws_r = ws;
    float* ws_i = ws + NTOT;
    const size_t base = (size_t)bt * 2560;
    for (int e = tid; e < 2560; e += 384) {
      int k = e >> 6, c = e & 63;
      float s = dsum[k];
      ws_r[base + e] = st_r[c*48 + k] * s;
      ws_i[base + e] = st_i[c*48 + k] * s;
    }
  }
}

// ---------------------------------------------------------------------------
// Kernel 2: first-order IIR over T. One thread per (comp, b, k, c) scalar lane
// (10240 lanes), sequential over T=400, in-place in ws.
// ---------------------------------------------------------------------------
__global__ __launch_bounds__(256) void pv_iir_kernel(
    const float* __restrict__ tau, float* __restrict__ ws)
{
  int gid = blockIdx.x * 256 + threadIdx.x;
  if (gid >= 10240) return;
  int comp = gid / 5120;
  int rem  = gid - comp * 5120;
  int b    = rem / 2560;
  int kc   = rem - b * 2560;
  int k    = kc >> 6;
  int c    = kc & 63;
  float a  = tau[k];
  float om = 1.f - a;
  float* p = ws + (size_t)comp * NTOT;
  size_t idx = ((size_t)(b * 400) * 40 + k) * 64 + c;
  float y = 0.f;
  for (int t = 0; t < 400; ++t) {
    float x = p[idx];
    y = a * y + om * x;
    p[idx] = y;
    idx += 2560;      // K*C2 stride per frame
  }
}

// ---------------------------------------------------------------------------
// Kernel 3: pv = Re( c2p x band_cov ) as WMMA GEMM [32000x64] x [64x64].
// 2000 M-tiles x 4 N-tiles; 8 waves per block.
// ---------------------------------------------------------------------------
__global__ __launch_bounds__(256) void pv_proj_kernel(
    const float* __restrict__ ws,
    const float* __restrict__ c2p_r, const float* __restrict__ c2p_i,
    float* __restrict__ out)
{
  const float* ws_r = ws;
  const float* ws_i = ws + NTOT;
  const int wg   = blockIdx.x * 8 + (threadIdx.x >> 5);
  const int mt   = wg >> 2;          // 0..1999
  const int nt   = wg & 3;           // 0..3
  const int lane = threadIdx.x & 31;
  const int ll   = lane & 15;
  const int hi   = lane >> 4;
  const int mrow = mt*16 + ll;       // flattened (b,t,k) row
  const int pcol = nt*16 + ll;       // power-vector index

  v8f acc = {};
  #pragma unroll 4
  for (int kk = 0; kk < 16; ++kk) {
    int kb = kk*4 + hi*2;
    v2f Ar, nAi, Br, Bi;
    Ar.x  = ws_r[(size_t)mrow*64 + kb];   Ar.y  = ws_r[(size_t)mrow*64 + kb + 1];
    nAi.x = -ws_i[(size_t)mrow*64 + kb];  nAi.y = -ws_i[(size_t)mrow*64 + kb + 1];
    Br.x  = c2p_r[pcol*64 + kb];          Br.y  = c2p_r[pcol*64 + kb + 1];
    Bi.x  = c2p_i[pcol*64 + kb];          Bi.y  = c2p_i[pcol*64 + kb + 1];
    acc = WMMA4(Ar,  Br, acc);     // + c2p_r * y_r
    acc = WMMA4(nAi, Bi, acc);     // - c2p_i * y_i  (real part only)
  }
  #pragma unroll
  for (int r = 0; r < 8; ++r) {
    int row = mt*16 + r + hi*8;
    int col = nt*16 + ll;
    out[(size_t)row*64 + col] = acc[r];
  }
}

// ---------------------------------------------------------------------------
extern "C" void kernel_launch(void* const* d_in, const int* in_sizes, int n_in,
                              void* d_out, int out_size, void* d_ws, size_t ws_size,
                              hipStream_t stream) {
  const float* bins_r = (const float*)d_in[0];
  const float* bins_i = (const float*)d_in[1];
  const float* bm_r   = (const float*)d_in[2];
  const float* bm_i   = (const float*)d_in[3];
  const float* c2p_r  = (const float*)d_in[4];
  const float* c2p_i  = (const float*)d_in[5];
  const float* tau    = (const float*)d_in[6];
  float* ws  = (float*)d_ws;
  float* out = (float*)d_out;

  const size_t lds_bytes = 23232 * sizeof(float);   // ~93 KB of the 320 KB WGP LDS
  pv_cov_band_kernel<<<dim3(NBT), dim3(384), lds_bytes, stream>>>(
      bins_r, bins_i, bm_r, bm_i, ws);
  pv_iir_kernel<<<dim3(40), dim3(256), 0, stream>>>(tau, ws);
  pv_proj_kernel<<<dim3(1000), dim3(256), 0, stream>>>(ws, c2p_r, c2p_i, out);
}